// GlobalQuantumAttention_38826504356540
// MI455X (gfx1250) — compile-verified
//
#include <hip/hip_runtime.h>
#include <cstdint>
#include <cstddef>

// Problem constants (match reference)
#define DIMD 64
#define BB   2
#define SS   512
#define TJ   32              // j-tile rows staged in LDS
#define JSPLIT 2             // j-range split across blocks (2 => bitwise-deterministic atomics)
#define NTB  (SS / TJ / JSPLIT)   // 8 tiles per block
#define WGSZ 256             // 8 wave32 per workgroup
#define TILE_FLOATS (TJ * DIMD)   // 2048 floats per array per tile
#define PLANE (BB * SS * DIMD)    // 65536 floats per (real/imag) output plane

// ---------------------------------------------------------------------------
// Kernel A: gp[b,d] = mean_i atan2(q_imag, q_real); store cos(gp), sin(gp).
// One wave32 per (b,d); each lane sums 16 strided rows, butterfly-reduce.
// ---------------------------------------------------------------------------
__global__ __launch_bounds__(32) void gqa_phase_mean(
    const float* __restrict__ qr, const float* __restrict__ qi,
    float* __restrict__ cg, float* __restrict__ sg) {
  const int blk  = blockIdx.x;          // 0..127 -> (b,d)
  const int b    = blk >> 6;
  const int d    = blk & (DIMD - 1);
  const int lane = threadIdx.x;         // 0..31

  const float* pr = qr + (size_t)b * SS * DIMD + d;
  const float* pi = qi + (size_t)b * SS * DIMD + d;
  float acc = 0.0f;
  #pragma unroll
  for (int k = 0; k < SS / 32; ++k) {   // 16 strided rows per lane
    const size_t i = (size_t)(lane + 32 * k) * DIMD;
    acc += atan2f(pi[i], pr[i]);
  }
  #pragma unroll
  for (int off = 16; off > 0; off >>= 1)
    acc += __shfl_xor(acc, off, 32);
  if (lane == 0) {
    const float gp = acc * (1.0f / (float)SS);
    cg[blk] = cosf(gp);
    sg[blk] = sinf(gp);
  }
}

// ---------------------------------------------------------------------------
// Issue the 12 per-lane async 16B copies that fill one 48KB tile buffer.
// GVS mode: mem = SGPR64 base + VGPR32 offset ; dest = LDS byte address.
// ---------------------------------------------------------------------------
__device__ __forceinline__ void issue_tile_loads(
    const float* const* bases, int b, int j0, const float* buf,
    unsigned goff0, unsigned goff1) {
  #pragma unroll
  for (int a = 0; a < 6; ++a) {
    const uint64_t gbase =
        (uint64_t)(uintptr_t)(bases[a] + ((size_t)b * SS + j0) * DIMD);
    const unsigned lds0 =
        (unsigned)(uintptr_t)(buf + a * TILE_FLOATS) + goff0;
    const unsigned lds1 = lds0 + (unsigned)(WGSZ * 16);
    asm volatile("global_load_async_to_lds_b128 %0, %1, %2"
                 :: "v"(lds0), "v"(goff0), "s"(gbase) : "memory");
    asm volatile("global_load_async_to_lds_b128 %0, %1, %2"
                 :: "v"(lds1), "v"(goff1), "s"(gbase) : "memory");
  }
}

// ---------------------------------------------------------------------------
// Main kernel: 256 blocks = (j-half, b, 8 i-rows). One wave per output row;
// lane owns dims (2*lane, 2*lane+1). Double-buffered CDNA5 async-LDS pipeline
// over 32-row j tiles; unrotated partial sums atomically added to d_out
// (exactly 2 adds per element onto zero => bitwise order-independent).
// ---------------------------------------------------------------------------
__global__ __launch_bounds__(WGSZ) void gqa_main(
    const float* __restrict__ qr, const float* __restrict__ qi,
    const float* __restrict__ kr, const float* __restrict__ ki,
    const float* __restrict__ vr, const float* __restrict__ vi,
    float* __restrict__ out) {
  // 2 buffers * 6 arrays * 32 rows * 64 f32 = 96KB (WGP pool is 320KB).
  // 16B alignment is REQUIRED: async b128 copies land on 16B LDS addresses.
  __shared__ __align__(16) float smem[2][6 * TILE_FLOATS];

  const int tid   = threadIdx.x;
  const int lane  = tid & 31;
  const int wave  = tid >> 5;
  const int blk   = blockIdx.x;                 // 0..255
  const int jhalf = blk >> 7;                   // 0..1
  const int inner = blk & 127;
  const int b     = inner >> 6;                 // 64 blocks per batch
  const int i     = ((inner & 63) << 3) + wave; // this wave's query row
  const int d0    = lane << 1;                  // lane owns d0, d0+1

  const float* bases[6] = {qr, qi, kr, ki, vr, vi};

  // This wave's q row values for its two dims.
  const size_t rowq = ((size_t)b * SS + i) * DIMD + d0;
  const float2 qri = *(const float2*)(qr + rowq);
  const float2 qii = *(const float2*)(qi + rowq);

  float accr0 = 0.f, accr1 = 0.f, acci0 = 0.f, acci1 = 0.f;
  const float invS  = 1.0f / (float)SS;
  const float scale = 0.125f;                   // 64^-0.5

  // Per-lane 16B chunks for the async fill: chunk tid and chunk tid+256.
  const unsigned goff0 = (unsigned)tid * 16u;
  const unsigned goff1 = goff0 + (unsigned)(WGSZ * 16);   // +4096

  const int t0 = jhalf * NTB;                   // this block's first tile

  // Prologue: fill buffer 0 with first tile of this block's j-range.
  issue_tile_loads(bases, b, t0 * TJ, smem[0], goff0, goff1);

  for (int t = 0; t < NTB; ++t) {
    const int cur = t & 1;
    if (t + 1 < NTB) {
      // Next tile into the other buffer (its last consumer finished at the
      // end-of-(t-1) barrier), then wait for *this* tile's 12 loads only:
      // ASYNCcnt completes in order, so <=12 outstanding == tile t landed.
      issue_tile_loads(bases, b, (t0 + t + 1) * TJ, smem[cur ^ 1], goff0, goff1);
      asm volatile("s_wait_asynccnt 12" ::: "memory");
    } else {
      asm volatile("s_wait_asynccnt 0" ::: "memory");
    }
    __syncthreads();

    const float* buf = smem[cur];
    for (int jj = 0; jj < TJ; ++jj) {
      const int e = jj * DIMD + d0;
      const float2 qrj = *(const float2*)(buf + 0 * TILE_FLOATS + e);
      const float2 qij = *(const float2*)(buf + 1 * TILE_FLOATS + e);
      const float2 krj = *(const float2*)(buf + 2 * TILE_FLOATS + e);
      const float2 kij = *(const float2*)(buf + 3 * TILE_FLOATS + e);
      const float2 vrj = *(const float2*)(buf + 4 * TILE_FLOATS + e);
      const float2 vij = *(const float2*)(buf + 5 * TILE_FLOATS + e);

      // fast phase diff + cubic correction, interference = cos(pd)/S.
      // IEEE divide + precise cosf: pd^3 can be enormous when dot ~ 0,
      // so only correct rounding + real range reduction tracks the reference.
      float dot0 = qri.x * qrj.x + qii.x * qij.x;
      float dot1 = qri.y * qrj.y + qii.y * qij.y;
      float cr0  = qii.x * qrj.x - qri.x * qij.x;
      float cr1  = qii.y * qrj.y - qri.y * qij.y;
      float pd0  = cr0 / (fabsf(dot0) + 1e-6f);
      float pd1  = cr1 / (fabsf(dot1) + 1e-6f);
      pd0 = pd0 * (1.0f - 0.28125f * pd0 * pd0);
      pd1 = pd1 * (1.0f - 0.28125f * pd1 * pd1);
      const float itf0 = cosf(pd0) * invS;
      const float itf1 = cosf(pd1) * invS;

      // phase of elementwise complex difference, scaled: |x| <= pi/8,
      // so softmax needs no max-subtraction (exp in [0.67, 1.48]).
      const float x0 = atan2f(qii.x - kij.x, qri.x - krj.x) * scale;
      const float x1 = atan2f(qii.y - kij.y, qri.y - krj.y) * scale;

      const float e0 = __expf(x0);
      const float e1 = __expf(x1);
      float s = e0 + e1;
      #pragma unroll
      for (int off = 16; off > 0; off >>= 1)
        s += __shfl_xor(s, off, 32);
      const float inv = __builtin_amdgcn_rcpf(s);   // v_rcp_f32, ~1 ulp

      const float w0 = e0 * inv * itf0;
      const float w1 = e1 * inv * itf1;
      accr0 += w0 * vrj.x;  accr1 += w1 * vrj.y;
      acci0 += w0 * vij.x;  acci1 += w1 * vij.y;
    }
    __syncthreads();   // tile fully consumed before its buffer is re-filled
  }

  // ---- Accumulate unrotated partials (2 contributions per element) ----
  const size_t obase = ((size_t)b * SS + i) * DIMD + d0;
  atomicAdd(out + obase,             accr0);
  atomicAdd(out + obase + 1,         accr1);
  atomicAdd(out + PLANE + obase,     acci0);
  atomicAdd(out + PLANE + obase + 1, acci1);
}

// ---------------------------------------------------------------------------
// Epilogue: in-place entanglement rotation across the two planes.
// Each thread owns one float2 of (b,i,d)-space exclusively.
// ---------------------------------------------------------------------------
__global__ __launch_bounds__(256) void gqa_rotate(
    const float* __restrict__ cg, const float* __restrict__ sg,
    float* __restrict__ out) {
  const int t = blockIdx.x * 256 + threadIdx.x;   // 0 .. PLANE/2-1
  const size_t obase = (size_t)t * 2;
  const int d0 = (int)(obase & (DIMD - 1));
  const int b  = (int)(obase / ((size_t)SS * DIMD));

  float2 r  = *(float2*)(out + obase);
  float2 im = *(float2*)(out + PLANE + obase);
  const float c0 = cg[b * DIMD + d0],  c1 = cg[b * DIMD + d0 + 1];
  const float s0 = sg[b * DIMD + d0],  s1 = sg[b * DIMD + d0 + 1];
  const float er0 = r.x * c0 - im.x * s0;
  const float er1 = r.y * c1 - im.y * s1;
  const float ei0 = r.x * s0 + im.x * c0;
  const float ei1 = r.y * s1 + im.y * c1;
  *(float2*)(out + obase)         = make_float2(er0, er1);
  *(float2*)(out + PLANE + obase) = make_float2(ei0, ei1);
}

// ---------------------------------------------------------------------------
extern "C" void kernel_launch(void* const* d_in, const int* in_sizes, int n_in,
                              void* d_out, int out_size, void* d_ws, size_t ws_size,
                              hipStream_t stream) {
  (void)in_sizes; (void)n_in; (void)ws_size;
  const float* qr = (const float*)d_in[0];
  const float* qi = (const float*)d_in[1];
  const float* kr = (const float*)d_in[2];
  const float* ki = (const float*)d_in[3];
  const float* vr = (const float*)d_in[4];
  const float* vi = (const float*)d_in[5];
  float* out = (float*)d_out;

  float* cg = (float*)d_ws;            // [B*D] cos(mean phase)
  float* sg = cg + BB * DIMD;          // [B*D] sin(mean phase)

  hipMemsetAsync(out, 0, (size_t)out_size * sizeof(float), stream);
  gqa_phase_mean<<<dim3(BB * DIMD), 32, 0, stream>>>(qr, qi, cg, sg);
  gqa_main<<<dim3(JSPLIT * BB * (SS / 8)), WGSZ, 0, stream>>>(qr, qi, kr, ki,
                                                              vr, vi, out);
  gqa_rotate<<<dim3(PLANE / 2 / 256), 256, 0, stream>>>(cg, sg, out);
}